// NUFFTForwardModel_73194832658721
// MI455X (gfx1250) — compile-verified
//
#include <hip/hip_runtime.h>
#include <math.h>
#include <stdint.h>

// Problem constants (match reference)
#define B_    8
#define NPIX  256
#define M_    8192
#define NCP   24576
#define NCA   24576
#define EPSF  1e-16f
#define MT    16                 // m-tile (WMMA M)
#define XTG   4                  // x-tiles processed per k-pass (register blocking)
#define WPB   4                  // waves (m-tiles) per block sharing the LDS slab
#define NBLK  (B_ * (M_/MT) / WPB)
#define SLABSTRIDE 72            // padded LDS row stride (dwords): 2*72 mod 64 = 16
                                 // -> half-waves (rows kk vs kk+2) hit disjoint banks

typedef __attribute__((ext_vector_type(2))) float v2f;
typedef __attribute__((ext_vector_type(4))) float v4f;
typedef __attribute__((ext_vector_type(8))) float v8f;

// ---------------------------------------------------------------------------
// Kernel 1: build DFT matrices.
//   Ev   [M][NPIX] (m-major)   : exp(-i*kv[m]*coord[p])
//   Eu_t [NPIX][M] (p-major)   : exp(-i*ku[m]*coord[p])  -- TRANSPOSED so the
//        GEMM epilogue can read 8 consecutive m's with b128 loads.
// Both store mappings are chosen so every global_store is coalesced.
// ---------------------------------------------------------------------------
__global__ void build_E_kernel(const float* __restrict__ ktraj,
                               float* __restrict__ Ev_r,  float* __restrict__ Ev_i,
                               float* __restrict__ Eu_rt, float* __restrict__ Eu_it) {
    int idx = blockIdx.x * blockDim.x + threadIdx.x;
    if (idx >= M_ * NPIX) return;
    float s, co;

    // Ev, m-major: consecutive threads -> consecutive p  (coalesced)
    {
        int m = idx / NPIX;
        int p = idx - m * NPIX;
        float c  = (float)(p - NPIX / 2);
        float kv = ktraj[m];            // ktraj[0][m]
        sincosf(kv * c, &s, &co);
        Ev_r[idx] = co; Ev_i[idx] = -s;
    }
    // Eu transposed, p-major: consecutive threads -> consecutive m (coalesced)
    {
        int p = idx / M_;
        int m = idx - p * M_;
        float c  = (float)(p - NPIX / 2);
        float ku = ktraj[M_ + m];       // ktraj[1][m]
        sincosf(ku * c, &s, &co);
        Eu_rt[idx] = co; Eu_it[idx] = -s;
    }
}

// ---------------------------------------------------------------------------
// Kernel 2: fused DFT via fp32 WMMA + async-to-LDS image staging.
// 4 waves per block; wave w owns m-tile blockIdx*4+w (same batch b).
//   tmp[m,x]  = sum_y Ev[m,y] * img[b,y,x]      (2 real GEMMs via v_wmma f32)
//   kdata[m]  = sum_x tmp[m,x] * Eu[m,x]        (fused epilogue + shfl reduce)
//   vis       = kdata * pf ;  visamp = |vis|
//
// The image B-fragments depend only on (y,x), so the 4 waves share one LDS
// slab (128 rows x 64 cols, stride 72) filled with
// GLOBAL_LOAD_ASYNC_TO_LDS_B128 (ASYNCcnt) -- 4x less L2 image traffic.
//
// C-matrix layout (ISA 7.12.2, 32-bit 16x16): lane l holds N = l%16,
// VGPR r holds row M = r + 8*(l/16).
// A 16x4 f32: lane l -> M=l%16; VGPR0={K0 lanes0-15, K2 lanes16-31}, VGPR1={K1,K3}.
// B 4x16 f32: lane l -> N=l%16; lanes0-15 hold K={0,1}, lanes16-31 K={2,3}.
// Uniform control flow -> EXEC all-1s around every WMMA as required.
// ---------------------------------------------------------------------------
__global__ __launch_bounds__(128) void dft_wmma_kernel(
    const float* __restrict__ images,
    const float* __restrict__ Ev_r,  const float* __restrict__ Ev_i,
    const float* __restrict__ Eu_rt, const float* __restrict__ Eu_it,
    const float* __restrict__ pulsefac,
    float* __restrict__ vis,      // (B, 2, M)
    float* __restrict__ visamp)   // (B, M)
{
    __shared__ float slab[128 * SLABSTRIDE];   // 36 KB

    const int tid  = threadIdx.x;
    const int wave = tid >> 5;
    const int lane = tid & 31;
    const int nloc = lane & 15;   // N position / A-row within tile
    const int half = lane >> 4;   // 0: K={0,1}, 1: K={2,3}

    const int tile = blockIdx.x * WPB + wave;         // global m-tile id
    const int b    = tile / (M_ / MT);                // same b for all 4 waves
    const int m0   = (tile - b * (M_ / MT)) * MT;

    const float* __restrict__ img = images + (size_t)b * NPIX * NPIX;
    const float* __restrict__ evr_row = Ev_r + (size_t)(m0 + nloc) * NPIX;
    const float* __restrict__ evi_row = Ev_i + (size_t)(m0 + nloc) * NPIX;

    float kr[8], ki[8];
#pragma unroll
    for (int r = 0; r < 8; ++r) { kr[r] = 0.0f; ki[r] = 0.0f; }

    for (int xg = 0; xg < NPIX; xg += 16 * XTG) {     // 4 groups of 4 x-tiles
        v8f cr[XTG], ci[XTG];
#pragma unroll
        for (int t = 0; t < XTG; ++t) { cr[t] = (v8f){}; ci[t] = (v8f){}; }

        for (int yh = 0; yh < 2; ++yh) {              // two 128-row halves of y
            const int y0 = yh * 128;

            __syncthreads();  // previous slab fully consumed by all waves

            // ---- async stage: img[y0..y0+127][xg..xg+63] -> LDS slab ----
            // 2048 b128 chunks, 16 per thread. 16 chunks per row of 64 floats.
#pragma unroll
            for (int it = 0; it < 16; ++it) {
                const int c   = it * 128 + tid;
                const int row = c >> 4;
                const int q   = c & 15;
                const float* gp = img + (size_t)(y0 + row) * NPIX + xg + q * 4;
                const uint64_t ga = (uint64_t)(uintptr_t)gp;
                // flat LDS aperture keeps the LDS byte offset in addr[31:0]
                const uint32_t lp = (uint32_t)(uintptr_t)&slab[row * SLABSTRIDE + q * 4];
                asm volatile("global_load_async_to_lds_b128 %0, %1, off"
                             :: "v"(lp), "v"(ga) : "memory");
            }
            asm volatile("s_wait_asynccnt 0" ::: "memory");  // this wave's DMA done
            __syncthreads();                                 // all waves' DMA done

            // ---- WMMA sweep over this y-half ----
            for (int k0 = 0; k0 < 128; k0 += 4) {
                const int kk = k0 + half * 2;                // local rows kk, kk+1
                v2f ar = *(const v2f*)(evr_row + y0 + kk);
                v2f ai = *(const v2f*)(evi_row + y0 + kk);
                const float* s0 = &slab[kk * SLABSTRIDE];
                const float* s1 = &slab[(kk + 1) * SLABSTRIDE];
#pragma unroll
                for (int t = 0; t < XTG; ++t) {
                    const int xo = t * 16 + nloc;
                    v2f bb;
                    bb.x = s0[xo];
                    bb.y = s1[xo];
                    // (neg_a, A, neg_b, B, c_mod, C, reuse_a, reuse_b)
                    cr[t] = __builtin_amdgcn_wmma_f32_16x16x4_f32(false, ar, false, bb,
                                                                  (short)0, cr[t], false, false);
                    ci[t] = __builtin_amdgcn_wmma_f32_16x16x4_f32(false, ai, false, bb,
                                                                  (short)0, ci[t], false, false);
                }
            }
        }

        // ---- fused Eu reduction: lane owns column x, rows m0 + r + 8*half.
        // Eu is transposed [NPIX][M], so rows r=0..7 are 8 consecutive floats.
#pragma unroll
        for (int t = 0; t < XTG; ++t) {
            const int x = xg + t * 16 + nloc;
            const size_t eb = (size_t)x * M_ + m0 + 8 * half;   // 32B aligned
            v4f eur_lo = *(const v4f*)(Eu_rt + eb);
            v4f eur_hi = *(const v4f*)(Eu_rt + eb + 4);
            v4f eui_lo = *(const v4f*)(Eu_it + eb);
            v4f eui_hi = *(const v4f*)(Eu_it + eb + 4);
#pragma unroll
            for (int r = 0; r < 8; ++r) {
                const float eur = (r < 4) ? eur_lo[r] : eur_hi[r - 4];
                const float eui = (r < 4) ? eui_lo[r] : eui_hi[r - 4];
                const float tr = cr[t][r], ti = ci[t][r];
                kr[r] += tr * eur - ti * eui;
                ki[r] += tr * eui + ti * eur;
            }
        }
    }

    // Reduce the 16 x-columns held by each half-wave (masks stay within halves)
#pragma unroll
    for (int r = 0; r < 8; ++r) {
        float a = kr[r], c = ki[r];
        a += __shfl_xor(a, 8, 32);  c += __shfl_xor(c, 8, 32);
        a += __shfl_xor(a, 4, 32);  c += __shfl_xor(c, 4, 32);
        a += __shfl_xor(a, 2, 32);  c += __shfl_xor(c, 2, 32);
        a += __shfl_xor(a, 1, 32);  c += __shfl_xor(c, 1, 32);
        kr[r] = a; ki[r] = c;
    }

    if (nloc == 0) {   // lane 0 -> rows m0..m0+7, lane 16 -> rows m0+8..m0+15
#pragma unroll
        for (int r = 0; r < 8; ++r) {
            const int m = m0 + r + 8 * half;
            const float pr = pulsefac[m];        // pulsefac[0][m]
            const float pi = pulsefac[M_ + m];   // pulsefac[1][m]
            const float vr = kr[r] * pr - ki[r] * pi;
            const float vi = kr[r] * pi + ki[r] * pr;
            vis[((size_t)b * 2 + 0) * M_ + m] = vr;
            vis[((size_t)b * 2 + 1) * M_ + m] = vi;
            visamp[(size_t)b * M_ + m] = sqrtf(vr * vr + vi * vi + EPSF);
        }
    }
}

// ---------------------------------------------------------------------------
// Kernel 3: closure phases.  cphase[b,n] = (180/pi) * sum_k sign[k,n]*atan2(...)
// ---------------------------------------------------------------------------
__global__ void cphase_kernel(const float* __restrict__ vis,
                              const float* __restrict__ sign,
                              const int*   __restrict__ ind,
                              float* __restrict__ out) {
    int i = blockIdx.x * blockDim.x + threadIdx.x;
    if (i >= B_ * NCP) return;
    int b = i / NCP;
    int n = i - b * NCP;
    const float* vr = vis + (size_t)b * 2 * M_;
    const float* vi = vr + M_;
    float acc = 0.0f;
#pragma unroll
    for (int k = 0; k < 3; ++k) {
        int idx = ind[k * NCP + n];
        acc += sign[k * NCP + n] * atan2f(vi[idx], vr[idx]);
    }
    out[i] = acc * 57.29577951308232f;   // 180/pi
}

// ---------------------------------------------------------------------------
// Kernel 4: log closure amplitudes.
// ---------------------------------------------------------------------------
__global__ void logcamp_kernel(const float* __restrict__ vis,
                               const int*   __restrict__ ind,
                               float* __restrict__ out) {
    int i = blockIdx.x * blockDim.x + threadIdx.x;
    if (i >= B_ * NCA) return;
    int b = i / NCA;
    int n = i - b * NCA;
    const float* vr = vis + (size_t)b * 2 * M_;
    const float* vi = vr + M_;
    float acc = 0.0f;
#pragma unroll
    for (int k = 0; k < 4; ++k) {
        int idx = ind[k * NCA + n];
        float rr = vr[idx], ii = vi[idx];
        float a = sqrtf(rr * rr + ii * ii + EPSF);
        float l = logf(a);
        acc += (k < 2) ? l : -l;
    }
    out[i] = acc;
}

// ---------------------------------------------------------------------------
extern "C" void kernel_launch(void* const* d_in, const int* in_sizes, int n_in,
                              void* d_out, int out_size, void* d_ws, size_t ws_size,
                              hipStream_t stream) {
    const float* images      = (const float*)d_in[0];  // (B, NPIX, NPIX)
    const float* ktraj       = (const float*)d_in[1];  // (2, M)
    const float* pulsefac    = (const float*)d_in[2];  // (2, M)
    const float* cphase_sign = (const float*)d_in[3];  // (3, NCP)
    const int*   cphase_ind  = (const int*)  d_in[4];  // (3, NCP)
    const int*   camp_ind    = (const int*)  d_in[5];  // (4, NCA)

    float* out     = (float*)d_out;
    float* vis     = out;                          // B*2*M   = 131072
    float* visamp  = vis    + (size_t)B_ * 2 * M_; // B*M     =  65536
    float* cphase  = visamp + (size_t)B_ * M_;     // B*NCP   = 196608
    float* logcamp = cphase + (size_t)B_ * NCP;    // B*NCA   = 196608

    // Workspace: 4 x (M*NPIX) floats = 33.5 MB
    float* Ev_r  = (float*)d_ws;
    float* Ev_i  = Ev_r  + (size_t)M_ * NPIX;
    float* Eu_rt = Ev_i  + (size_t)M_ * NPIX;
    float* Eu_it = Eu_rt + (size_t)M_ * NPIX;

    {
        int n = M_ * NPIX;
        build_E_kernel<<<(n + 255) / 256, 256, 0, stream>>>(ktraj, Ev_r, Ev_i, Eu_rt, Eu_it);
    }

    dft_wmma_kernel<<<NBLK, 128, 0, stream>>>(images, Ev_r, Ev_i, Eu_rt, Eu_it,
                                              pulsefac, vis, visamp);

    {
        int n = B_ * NCP;
        cphase_kernel<<<(n + 255) / 256, 256, 0, stream>>>(vis, cphase_sign, cphase_ind, cphase);
    }
    {
        int n = B_ * NCA;
        logcamp_kernel<<<(n + 255) / 256, 256, 0, stream>>>(vis, camp_ind, logcamp);
    }
}